// CTCLayer_15676630630523
// MI455X (gfx1250) — compile-verified
//
#include <hip/hip_runtime.h>
#include <stdint.h>

// ---------------------------------------------------------------------------
// CTC loss (keras ctc_batch_cost semantics), MI455X / gfx1250.
// Dims fixed by the reference setup: B=128, T=512, C=1024, U=128.
//
// Roofline: only y_hat (268 MB) matters -> one full-rate HBM pass ~11.5 us.
// Phase 1 (bandwidth-bound, massively parallel): gather
//   p[b,t,s] = y_hat[b,t,ext[b,s]] + eps  into d_ws lattice [B,T,320]
//   (~80 MB; needs ws_size >= 83,886,080 bytes; lattice stays L2-resident).
// Phase 2 (serial scan, one wave32 per batch element): rescaled
//   probability-domain CTC forward recursion (no per-step transcendentals;
//   renormalize every 8 steps, fold scales into logZ). Emissions streamed
//   through a 16-slot LDS ring via gfx1250 async global->LDS DMA
//   (global_load_async_to_lds_b64 / s_wait_asynccnt), 10 rows in flight.
// Lane-major state map s = lane*10 + r keeps alpha[s-1], alpha[s-2]
// in-register: only 2 cross-lane shuffles per time step.
// ---------------------------------------------------------------------------

#define CTC_B   128
#define CTC_T   512
#define CTC_C   1024
#define CTC_U   128
#define CTC_S   (2 * CTC_U + 1)   // 257 lattice states
#define CTC_SP  320               // padded: 32 lanes * 10 states/lane
#define RPL     10                // states per lane (lane-major: s = lane*RPL + r)
#define RING    16                // LDS ring slots (rows of CTC_SP floats)
#define AHEAD   10                // rows in flight (10 rows * 5 ops = 50 <= 63)
#define OPS_ROW (CTC_SP / 64)     // 5 async b64 ops per row (each moves 256B)
#define CTC_EPS 1e-7f

// ---- gfx1250 async global->LDS copy + ASYNCcnt wait (guarded) -------------
typedef int ctc_v2i __attribute__((ext_vector_type(2)));
typedef __attribute__((address_space(1))) ctc_v2i ctc_gv2i;  // global int2
typedef __attribute__((address_space(3))) ctc_v2i ctc_lv2i;  // LDS int2

#if __has_builtin(__builtin_amdgcn_global_load_async_to_lds_b64)
__device__ __forceinline__ void async_copy_b64(const float* g, float* l) {
  __builtin_amdgcn_global_load_async_to_lds_b64(
      (ctc_gv2i*)g, (ctc_lv2i*)l, 0, 0);
}
#elif defined(__gfx1250__)
__device__ __forceinline__ void async_copy_b64(const float* g, float* l) {
  uint32_t lds_addr = (uint32_t)(uintptr_t)l;       // flat LDS addr truncates to b32
  uint64_t gaddr    = (uint64_t)(uintptr_t)g;
  asm volatile("global_load_async_to_lds_b64 %0, %1, off"
               :: "v"(lds_addr), "v"(gaddr) : "memory");
}
#else
__device__ __forceinline__ void async_copy_b64(const float* g, float* l) {
  *(float2*)l = *(const float2*)g;                  // host-pass / fallback
}
#endif

#if __has_builtin(__builtin_amdgcn_s_wait_asynccnt)
#define WAIT_ASYNC(n) do { __builtin_amdgcn_s_wait_asynccnt(n); \
                           asm volatile("" ::: "memory"); } while (0)
#elif defined(__gfx1250__)
#define WAIT_ASYNC(n) asm volatile("s_wait_asynccnt %0" :: "i"(n) : "memory")
#else
#define WAIT_ASYNC(n) do { } while (0)
#endif

// ---------------------------------------------------------------------------
// Phase 1: lattice gather. One block per (b,t) row; thread s writes
// p[b,t,s] = y_hat[b,t,ext[b,s]] + eps  (ext: blank=C-1 at even s, labels odd).
// Streams the 268 MB y_hat exactly once at HBM rate.
// ---------------------------------------------------------------------------
__global__ __launch_bounds__(CTC_SP) void ctc_gather(
    const float* __restrict__ yhat, const int* __restrict__ labels,
    float* __restrict__ rows) {
  const int s  = threadIdx.x;
  const int bt = blockIdx.x;          // b*T + t
  const int b  = bt >> 9;             // T = 512
  float v = 0.0f;                     // states >= S are dead padding (prob 0)
  if (s < CTC_S) {
    const int c = (s & 1) ? labels[b * CTC_U + (s >> 1)] : (CTC_C - 1);
    v = yhat[(size_t)bt * CTC_C + c] + CTC_EPS;
  }
  rows[(size_t)bt * CTC_SP + s] = v;
}

// ---------------------------------------------------------------------------
// Phase 2: rescaled prob-domain CTC forward. One wave32 per batch element.
// ---------------------------------------------------------------------------
__global__ __launch_bounds__(32) void ctc_alpha(
    const float* __restrict__ rows_all, const int* __restrict__ labels,
    float* __restrict__ out) {
  __shared__ __align__(16) float ring[RING * CTC_SP];   // 20 KB
  const int b    = blockIdx.x;
  const int lane = threadIdx.x;
  const float* rows = rows_all + (size_t)b * CTC_T * CTC_SP;

  // skip-transition mask: s odd, s>=3, labels[u] != labels[u-1]
  float skipf[RPL];
#pragma unroll
  for (int r = 0; r < RPL; ++r) {
    const int s = lane * RPL + r;
    float f = 0.0f;
    if ((s & 1) && s >= 3 && s < CTC_S) {
      const int u = (s - 1) >> 1;
      f = (labels[b * CTC_U + u] != labels[b * CTC_U + u - 1]) ? 1.0f : 0.0f;
    }
    skipf[r] = f;
  }

  // alpha(t=0): only states 0 (blank) and 1 (first label), both on lane 0
  float alpha[RPL];
#pragma unroll
  for (int r = 0; r < RPL; ++r) alpha[r] = 0.0f;
  {
    const float p0 = rows[0];
    const float p1 = rows[1];
    if (lane == 0) { alpha[0] = p0; alpha[1] = p1; }
  }

  auto issue_row = [&](int tp) {
    const float* g = rows + (size_t)tp * CTC_SP;
    float* l = &ring[(tp & (RING - 1)) * CTC_SP];
#pragma unroll
    for (int k = 0; k < OPS_ROW; ++k)
      async_copy_b64(g + (k * 32 + lane) * 2, l + (k * 32 + lane) * 2);
  };

  // prologue: prefetch rows 1..AHEAD into the LDS ring (async DMA)
  for (int tp = 1; tp <= AHEAD; ++tp) issue_row(tp);

  float logZ = 0.0f;

  auto step = [&](int tt, bool rescale) {
    // emissions for this step from the ring (b64 LDS loads, 8B-aligned)
    const float2* p2 =
        (const float2*)&ring[(tt & (RING - 1)) * CTC_SP + lane * RPL];
    float pv[RPL];
#pragma unroll
    for (int k = 0; k < RPL / 2; ++k) {
      const float2 q = p2[k];
      pv[2 * k] = q.x; pv[2 * k + 1] = q.y;
    }
    // cross-lane carries: states lane*RPL-1 and lane*RPL-2
    float carry1 = __shfl_up(alpha[RPL - 1], 1);
    float carry2 = __shfl_up(alpha[RPL - 2], 1);
    if (lane == 0) { carry1 = 0.0f; carry2 = 0.0f; }
    // in-place descending update (reads old lower-index alphas)
#pragma unroll
    for (int r = RPL - 1; r >= 2; --r)
      alpha[r] = (alpha[r] + alpha[r - 1] + skipf[r] * alpha[r - 2]) * pv[r];
    alpha[1] = (alpha[1] + alpha[0] + skipf[1] * carry1) * pv[1];
    alpha[0] = (alpha[0] + carry1 + skipf[0] * carry2) * pv[0];
    // renormalize; fold scale into logZ (alpha decays ~1e-21 per 8 steps,
    // safely inside f32 range between rescales)
    if (rescale) {
      float ssum = 0.0f;
#pragma unroll
      for (int r = 0; r < RPL; ++r) ssum += alpha[r];
#pragma unroll
      for (int off = 16; off >= 1; off >>= 1) ssum += __shfl_xor(ssum, off);
      const float inv = 1.0f / ssum;
      logZ += logf(ssum);
#pragma unroll
      for (int r = 0; r < RPL; ++r) alpha[r] *= inv;
    }
  };

  // steady state: 62 groups of 8 steps, t = 1..496.
  // Max prefetch row = 496+10 = 506 <= 511; in-flight <= 11 rows (55 ops).
  // Rescale flag folds to a constant per unrolled step (k==7 <=> t%8==0).
  int t = 1;
  for (int g = 0; g < 62; ++g) {
#pragma unroll
    for (int k = 0; k < 8; ++k, ++t) {
      issue_row(t + AHEAD);
      WAIT_ASYNC(OPS_ROW * AHEAD);             // row t complete (<=50 left)
      step(t, k == 7);
    }
  }
  // tail: t = 497..511 (issue while rows remain, then drain)
  for (; t <= CTC_T - 1; ++t) {
    if (t + AHEAD <= CTC_T - 1) {
      issue_row(t + AHEAD);
      WAIT_ASYNC(OPS_ROW * AHEAD);
    } else {
      WAIT_ASYNC(0);
    }
    step(t, (t & 7) == 0);
  }

  // loss = -(log(alpha[S-1] + alpha[S-2]) + logZ); states 255,256 on lane 25
  const float tail = __shfl(alpha[5] + alpha[6], 25);
  if (lane == 0) out[b] = -(logf(tail) + logZ);
}

// ---------------------------------------------------------------------------
extern "C" void kernel_launch(void* const* d_in, const int* in_sizes, int n_in,
                              void* d_out, int out_size, void* d_ws,
                              size_t ws_size, hipStream_t stream) {
  const int*   y_true = (const int*)d_in[0];    // [B, U] int32
  const float* y_hat  = (const float*)d_in[1];  // [B, T, C] f32
  float* out  = (float*)d_out;                  // [B] f32
  float* rows = (float*)d_ws;                   // [B, T, 320] f32 (80 MB)

  ctc_gather<<<CTC_B * CTC_T, CTC_SP, 0, stream>>>(y_hat, y_true, rows);
  ctc_alpha<<<CTC_B, 32, 0, stream>>>(rows, y_true, out);

  (void)in_sizes; (void)n_in; (void)out_size; (void)ws_size;
}